// ChannelSelection_Top_73804718014604
// MI455X (gfx1250) — compile-verified
//
#include <hip/hip_runtime.h>
#include <hip/hip_bf16.h>

typedef __attribute__((ext_vector_type(2))) float v2f;
typedef __attribute__((ext_vector_type(8))) float v8f;

#define BN_EPS 1e-5f

// ---------------------------------------------------------------------------
// Pass 1: per-(b,c) plane mean.  One block (256 threads) per plane.
// ---------------------------------------------------------------------------
__global__ void plane_mean_kernel(const float* __restrict__ x,
                                  float* __restrict__ y, int HW) {
  const int plane = blockIdx.x;
  const float* p = x + (size_t)plane * (size_t)HW;
  float s = 0.f;
  for (int i = threadIdx.x; i < HW; i += 256) s += p[i];
  __shared__ float red[256];
  red[threadIdx.x] = s;
  __syncthreads();
  for (int off = 128; off > 0; off >>= 1) {
    if ((int)threadIdx.x < off) red[threadIdx.x] += red[threadIdx.x + off];
    __syncthreads();
  }
  if (threadIdx.x == 0) y[plane] = red[0] / (float)HW;
}

// ---------------------------------------------------------------------------
// Pass 2: SE MLP + sigmoid + stable top-128 (rank by count, ties -> lower idx,
// matching jax.lax.top_k).  One block (256 threads) per batch.
// ---------------------------------------------------------------------------
template <int C, int H>
__global__ void se_topk_kernel(const float* __restrict__ means,
                               const float* __restrict__ w1,   // [H, C]
                               const float* __restrict__ w2,   // [C, H]
                               int* __restrict__ idx_out) {    // [B, 128]
  const int b = blockIdx.x;
  __shared__ float sy[C];
  __shared__ float sh[H];
  __shared__ float ss[C];
  for (int c = threadIdx.x; c < C; c += 256) sy[c] = means[b * C + c];
  __syncthreads();
  for (int h = threadIdx.x; h < H; h += 256) {
    float acc = 0.f;
    for (int c = 0; c < C; ++c) acc += w1[h * C + c] * sy[c];
    sh[h] = fmaxf(acc, 0.f);
  }
  __syncthreads();
  for (int c = threadIdx.x; c < C; c += 256) {
    float acc = 0.f;
    for (int h = 0; h < H; ++h) acc += w2[c * H + h] * sh[h];
    ss[c] = 1.f / (1.f + expf(-acc));
  }
  __syncthreads();
  for (int c = threadIdx.x; c < C; c += 256) {
    const float v = ss[c];
    int rank = 0;
    for (int c2 = 0; c2 < C; ++c2) {
      const float v2 = ss[c2];
      rank += (v2 > v) || ((v2 == v) && (c2 < c));
    }
    if (rank < 128) idx_out[b * 128 + rank] = c;
  }
}

// ---------------------------------------------------------------------------
// Pass 3: fused (resize -> gather top-128 channels -> 128x128 GEMM via WMMA
// f32 16x16x4 -> BN -> ReLU -> write into concat slice).
// Grid: (100 tiles of 64 spatial, 16 batches).  Block: 256 threads = 8 waves.
// Wave w owns M-rows [16w,16w+16); iterates 4 N-subtiles of the 64-wide tile.
// MODE 0: 2x2 avg pool from 160x160 (C=256)
// MODE 2: 2x bilinear (half-pixel) from 40x40 (C=512)
// MODE 3: 4x bilinear (half-pixel) from 20x20 (C=1024)
// ---------------------------------------------------------------------------
template <int MODE>
__global__ void fused_gemm_kernel(const float* __restrict__ x,
                                  const int* __restrict__ sel,   // [B,128]
                                  const float* __restrict__ cw,  // [128,128]
                                  const float* __restrict__ bng,
                                  const float* __restrict__ bnb,
                                  const float* __restrict__ bnm,
                                  const float* __restrict__ bnv,
                                  float* __restrict__ out,       // [B,384,80,80]
                                  int obase) {
  constexpr int C  = (MODE == 0) ? 256 : (MODE == 2 ? 512 : 1024);
  constexpr int IN = (MODE == 0) ? 160 : (MODE == 2 ? 40 : 20);

  const int b = blockIdx.y;
  const int ptile = blockIdx.x * 64;

  __shared__ int   s_idx[128];
  __shared__ float Xs[128 * 64];   // 32 KB: [K=128][N=64] gathered+resized

  if (threadIdx.x < 128) s_idx[threadIdx.x] = sel[b * 128 + threadIdx.x];
  __syncthreads();

  // Stage B-matrix tile: resize on the fly, no global materialization.
  for (int i = threadIdx.x; i < 128 * 64; i += 256) {
    const int k = i >> 6;
    const int n = i & 63;
    const int c = s_idx[k];
    const int p = ptile + n;
    const int oy = p / 80;
    const int ox = p - oy * 80;
    const float* pl = x + ((size_t)(b * C + c)) * (size_t)(IN * IN);
    float v;
    if (MODE == 0) {
      const int iy = oy * 2, ix = ox * 2;
      v = 0.25f * (pl[iy * 160 + ix] + pl[iy * 160 + ix + 1] +
                   pl[(iy + 1) * 160 + ix] + pl[(iy + 1) * 160 + ix + 1]);
    } else {
      const float scl = (MODE == 2) ? 0.5f : 0.25f;
      const float shf = (MODE == 2) ? 0.25f : 0.375f;
      const float sy = scl * (float)oy - shf;
      const float sx = scl * (float)ox - shf;
      int iy0 = (int)floorf(sy);
      int ix0 = (int)floorf(sx);
      const float fy = sy - (float)iy0;
      const float fx = sx - (float)ix0;
      const int iy1 = min(iy0 + 1, IN - 1); iy0 = max(iy0, 0);
      const int ix1 = min(ix0 + 1, IN - 1); ix0 = max(ix0, 0);
      const float v00 = pl[iy0 * IN + ix0], v01 = pl[iy0 * IN + ix1];
      const float v10 = pl[iy1 * IN + ix0], v11 = pl[iy1 * IN + ix1];
      v = (1.f - fy) * ((1.f - fx) * v00 + fx * v01) +
          fy * ((1.f - fx) * v10 + fx * v11);
    }
    Xs[i] = v;
  }
  __syncthreads();

  const int lane  = threadIdx.x & 31;
  const int wv    = threadIdx.x >> 5;
  const int mtile = wv * 16;
  const int mlane = lane & 15;
  const int khalf = (lane >> 4) << 1;  // lanes 0-15 -> K {0,1}; 16-31 -> {2,3}
  const int m     = mtile + mlane;

  v8f acc[4] = {};

  for (int kk = 0; kk < 128; kk += 4) {
    // A fragment (weights, M x K = 16x4) straight from global (L2 resident).
    v2f a;
    a.x = cw[m * 128 + kk + khalf];
    a.y = cw[m * 128 + kk + khalf + 1];
#pragma unroll
    for (int ns = 0; ns < 4; ++ns) {
      // B fragment (K x N = 4x16) from LDS.
      const int n = ns * 16 + mlane;
      v2f bb;
      bb.x = Xs[(kk + khalf) * 64 + n];
      bb.y = Xs[(kk + khalf + 1) * 64 + n];
      acc[ns] = __builtin_amdgcn_wmma_f32_16x16x4_f32(
          /*neg_a=*/false, a, /*neg_b=*/false, bb,
          /*c_mod=*/(short)0, acc[ns], /*reuse_a=*/false, /*reuse_b=*/false);
    }
  }

  // Epilogue: BN affine + ReLU, write into concat slice.
  // C/D layout: VGPR r -> rows (mtile+r) for lanes 0-15, (mtile+r+8) for 16-31.
  const int rhalf = (lane >> 4) << 3;
#pragma unroll
  for (int r = 0; r < 8; ++r) {
    const int row = mtile + r + rhalf;
    const float scale = bng[row] * rsqrtf(bnv[row] + BN_EPS);
    const float shift = bnb[row] - bnm[row] * scale;
#pragma unroll
    for (int ns = 0; ns < 4; ++ns) {
      const int col = ptile + ns * 16 + mlane;
      float vv = acc[ns][r] * scale + shift;
      vv = fmaxf(vv, 0.f);
      out[((size_t)(b * 384 + obase + row)) * 6400 + col] = vv;
    }
  }
}

// ---------------------------------------------------------------------------
extern "C" void kernel_launch(void* const* d_in, const int* in_sizes, int n_in,
                              void* d_out, int out_size, void* d_ws, size_t ws_size,
                              hipStream_t stream) {
  (void)in_sizes; (void)n_in; (void)out_size; (void)ws_size;

  const float* x0 = (const float*)d_in[0];
  const float* x2 = (const float*)d_in[2];
  const float* x3 = (const float*)d_in[3];

  const float* se0_w1 = (const float*)d_in[4];
  const float* se0_w2 = (const float*)d_in[5];
  const float* c0w    = (const float*)d_in[6];
  const float* bn0g   = (const float*)d_in[7];
  const float* bn0b   = (const float*)d_in[8];
  const float* bn0m   = (const float*)d_in[9];
  const float* bn0v   = (const float*)d_in[10];

  const float* se2_w1 = (const float*)d_in[11];
  const float* se2_w2 = (const float*)d_in[12];
  const float* c2w    = (const float*)d_in[13];
  const float* bn2g   = (const float*)d_in[14];
  const float* bn2b   = (const float*)d_in[15];
  const float* bn2m   = (const float*)d_in[16];
  const float* bn2v   = (const float*)d_in[17];

  const float* se3_w1 = (const float*)d_in[18];
  const float* se3_w2 = (const float*)d_in[19];
  const float* c3w    = (const float*)d_in[20];
  const float* bn3g   = (const float*)d_in[21];
  const float* bn3b   = (const float*)d_in[22];
  const float* bn3m   = (const float*)d_in[23];
  const float* bn3v   = (const float*)d_in[24];

  float* out = (float*)d_out;

  // Workspace layout (tiny: ~140 KB)
  float* mean0 = (float*)d_ws;            // 16*256
  float* mean2 = mean0 + 16 * 256;        // 16*512
  float* mean3 = mean2 + 16 * 512;        // 16*1024
  int*   idx0  = (int*)(mean3 + 16 * 1024);
  int*   idx2  = idx0 + 16 * 128;
  int*   idx3  = idx2 + 16 * 128;

  // Pass 1: channel means (resize-invariant, see analysis).
  plane_mean_kernel<<<16 * 256, 256, 0, stream>>>(x0, mean0, 160 * 160);
  plane_mean_kernel<<<16 * 512, 256, 0, stream>>>(x2, mean2, 40 * 40);
  plane_mean_kernel<<<16 * 1024, 256, 0, stream>>>(x3, mean3, 20 * 20);

  // Pass 2: SE scores + top-128 channel indices per batch.
  se_topk_kernel<256, 16><<<16, 256, 0, stream>>>(mean0, se0_w1, se0_w2, idx0);
  se_topk_kernel<512, 32><<<16, 256, 0, stream>>>(mean2, se2_w1, se2_w2, idx2);
  se_topk_kernel<1024, 64><<<16, 256, 0, stream>>>(mean3, se3_w1, se3_w2, idx3);

  // Pass 3: fused resize+gather+WMMA GEMM+BN+ReLU into concat output.
  dim3 grid(100, 16);
  fused_gemm_kernel<0><<<grid, 256, 0, stream>>>(x0, idx0, c0w, bn0g, bn0b, bn0m, bn0v, out, 0);
  fused_gemm_kernel<2><<<grid, 256, 0, stream>>>(x2, idx2, c2w, bn2g, bn2b, bn2m, bn2v, out, 128);
  fused_gemm_kernel<3><<<grid, 256, 0, stream>>>(x3, idx3, c3w, bn3g, bn3b, bn3m, bn3v, out, 256);
}